// ISTA_45964740002423
// MI455X (gfx1250) — compile-verified
//
#include <hip/hip_runtime.h>
#include <hip/hip_bf16.h>

#define BS_TOTAL 72
#define G 512
#define M2 64
#define LAYERS 16
#define POWER_ITERS 48
#define EPSF 1e-20f

typedef float v2f __attribute__((ext_vector_type(2)));
typedef float v8f __attribute__((ext_vector_type(8)));

// ---- workspace layout (float offsets) ----
constexpr size_t OFF_RE     = 0;                                   // 72*512*512
constexpr size_t OFF_IM     = OFF_RE + (size_t)BS_TOTAL * G * G;   // 72*512*512
constexpr size_t OFF_DHR_RE = OFF_IM + (size_t)BS_TOTAL * G * G;   // 72*512
constexpr size_t OFF_DHR_IM = OFF_DHR_RE + (size_t)BS_TOTAL * G;
constexpr size_t OFF_RES0   = OFF_DHR_IM + (size_t)BS_TOTAL * G;
constexpr size_t OFF_LAM    = OFF_RES0 + (size_t)BS_TOTAL * G;
constexpr size_t OFF_GAMMA  = OFF_LAM + BS_TOTAL;
constexpr size_t OFF_THETA  = OFF_GAMMA + BS_TOTAL;
constexpr size_t OFF_RESF   = OFF_THETA + BS_TOTAL;                // 72*512

// ---- output layout (floats): result | result_ave | result_init_all ----
constexpr size_t OUT_AVE  = (size_t)BS_TOTAL * G;        // 36864
constexpr size_t OUT_INIT = OUT_AVE + (size_t)8 * G;     // 40960

#define WMMA(A, B, C) \
  __builtin_amdgcn_wmma_f32_16x16x4_f32(false, (A), false, (B), (short)0, (C), false, false)

// ================================================================
// Kernel 1: DhD = conj(D)^T @ D  via V_WMMA_F32_16X16X4_F32
//   Re = Dr^T Dr + Di^T Di ; Im = Dr^T Di - Di^T Dr
// One wave -> one 16(i) x 32(j) complex output strip.
// ================================================================
__global__ __launch_bounds__(256) void dhd_wmma_kernel(
    const float* __restrict__ Dr, const float* __restrict__ Di,
    float* __restrict__ ws) {
  const unsigned wave = (blockIdx.x * 256u + threadIdx.x) >> 5;
  const unsigned lane = threadIdx.x & 31u;
  const unsigned jt = wave & 15u;          // 16 strips of 32 cols
  const unsigned it = (wave >> 4) & 31u;   // 32 tiles of 16 rows
  const unsigned bs = wave >> 9;           // 0..71
  if (bs >= BS_TOTAL) return;

  const float* __restrict__ dr = Dr + (size_t)bs * M2 * G;
  const float* __restrict__ di = Di + (size_t)bs * M2 * G;
  const int I0 = it * 16, J0 = jt * 32;
  const int m  = lane & 15;                // M (for A) / N (for B)
  const int kh = lane >> 4;                // K half-select

  v8f re0 = {}, re1 = {}, imA0 = {}, imB0 = {}, imA1 = {}, imB1 = {};
  for (int k = 0; k < M2; k += 4) {
    const int r0 = (k + 2 * kh) * G;
    const int r1 = r0 + G;
    v2f ar, ai, br0, bi0, br1, bi1;
    ar.x  = dr[r0 + I0 + m];       ar.y  = dr[r1 + I0 + m];
    ai.x  = di[r0 + I0 + m];       ai.y  = di[r1 + I0 + m];
    br0.x = dr[r0 + J0 + m];       br0.y = dr[r1 + J0 + m];
    bi0.x = di[r0 + J0 + m];       bi0.y = di[r1 + J0 + m];
    br1.x = dr[r0 + J0 + 16 + m];  br1.y = dr[r1 + J0 + 16 + m];
    bi1.x = di[r0 + J0 + 16 + m];  bi1.y = di[r1 + J0 + 16 + m];

    re0  = WMMA(ar, br0, re0);
    re0  = WMMA(ai, bi0, re0);
    imA0 = WMMA(ar, bi0, imA0);
    imB0 = WMMA(ai, br0, imB0);
    re1  = WMMA(ar, br1, re1);
    re1  = WMMA(ai, bi1, re1);
    imA1 = WMMA(ar, bi1, imA1);
    imB1 = WMMA(ai, br1, imB1);
  }

  float* __restrict__ Reo = ws + OFF_RE + (size_t)bs * G * G;
  float* __restrict__ Imo = ws + OFF_IM + (size_t)bs * G * G;
#pragma unroll
  for (int v = 0; v < 8; ++v) {
    const int row = I0 + v + 8 * kh;       // C/D layout: VGPR v -> M=v / M=v+8
    const size_t o0 = (size_t)row * G + J0 + m;
    Reo[o0]      = re0[v];
    Reo[o0 + 16] = re1[v];
    Imo[o0]      = imA0[v] - imB0[v];
    Imo[o0 + 16] = imA1[v] - imB1[v];
  }
}

// ---- block reduction helpers ----
__device__ __forceinline__ float blk_sum(float v, float* buf) {
  const int t = threadIdx.x;
  buf[t] = v; __syncthreads();
  for (int o = blockDim.x >> 1; o > 0; o >>= 1) {
    if (t < o) buf[t] += buf[t + o];
    __syncthreads();
  }
  const float r = buf[0]; __syncthreads();
  return r;
}
__device__ __forceinline__ float blk_max(float v, float* buf) {
  const int t = threadIdx.x;
  buf[t] = v; __syncthreads();
  for (int o = blockDim.x >> 1; o > 0; o >>= 1) {
    if (t < o) buf[t] = fmaxf(buf[t], buf[t + o]);
    __syncthreads();
  }
  const float r = buf[0]; __syncthreads();
  return r;
}
__device__ __forceinline__ float blk_min(float v, float* buf) {
  const int t = threadIdx.x;
  buf[t] = v; __syncthreads();
  for (int o = blockDim.x >> 1; o > 0; o >>= 1) {
    if (t < o) buf[t] = fminf(buf[t], buf[t + o]);
    __syncthreads();
  }
  const float r = buf[0]; __syncthreads();
  return r;
}

// ================================================================
// Kernel 2: per (b,s): Dhr = Dh @ r, res0 = normalize(Re(Dhr)),
//           lam = 0.5 * max|Dhr|
// r[k] = C[k%8][k/8]  (covariance transposed then flattened)
// ================================================================
__global__ __launch_bounds__(256) void setup_kernel(
    const float* __restrict__ Dr, const float* __restrict__ Di,
    const float* __restrict__ Cr, const float* __restrict__ Ci,
    float* __restrict__ ws) {
  __shared__ float rre[M2], rim[M2];
  __shared__ float buf[256];
  const int bs = blockIdx.x;
  const int t  = threadIdx.x;
  const float* __restrict__ dr = Dr + (size_t)bs * M2 * G;
  const float* __restrict__ di = Di + (size_t)bs * M2 * G;
  if (t < M2) {
    const int i = t >> 3, j = t & 7;       // r[i*8+j] = C[j][i]
    rre[t] = Cr[(size_t)bs * 64 + j * 8 + i];
    rim[t] = Ci[(size_t)bs * 64 + j * 8 + i];
  }
  __syncthreads();

  float sr[2], si[2];
  float nrm = 0.f, mx = 0.f;
#pragma unroll
  for (int p = 0; p < 2; ++p) {
    const int g = t + p * 256;
    float ar = 0.f, ai = 0.f;
    for (int k = 0; k < M2; ++k) {
      const float a = dr[(size_t)k * G + g];
      const float b = di[(size_t)k * G + g];
      ar += a * rre[k] + b * rim[k];       // Re(conj(D)*r)
      ai += a * rim[k] - b * rre[k];       // Im(conj(D)*r)
    }
    sr[p] = ar; si[p] = ai;
    nrm += ar * ar;
    mx = fmaxf(mx, sqrtf(ar * ar + ai * ai));
  }
  const float tot  = blk_sum(nrm, buf);
  const float mmax = blk_max(mx, buf);
  const float inv  = 1.f / (sqrtf(tot) + EPSF);
#pragma unroll
  for (int p = 0; p < 2; ++p) {
    const int g = t + p * 256;
    ws[OFF_DHR_RE + (size_t)bs * G + g] = sr[p];
    ws[OFF_DHR_IM + (size_t)bs * G + g] = si[p];
    ws[OFF_RES0   + (size_t)bs * G + g] = sr[p] * inv;
  }
  if (t == 0) ws[OFF_LAM + bs] = 0.5f * mmax;
}

// ================================================================
// Kernel 3: power iteration for lambda_max of Hermitian PSD DhD
//           (replaces eigvalsh: only max |eig| is needed)
// ================================================================
__global__ __launch_bounds__(512) void power_kernel(float* __restrict__ ws) {
  __shared__ float xr[G], xi[G], buf[512];
  const int bs = blockIdx.x, g = threadIdx.x;
  const float* __restrict__ Rm = ws + OFF_RE + (size_t)bs * G * G + (size_t)g * G;
  const float* __restrict__ Im = ws + OFF_IM + (size_t)bs * G * G + (size_t)g * G;

  float vr = ws[OFF_DHR_RE + (size_t)bs * G + g];
  float vi = ws[OFF_DHR_IM + (size_t)bs * G + g];
  const float n0 = blk_sum(vr * vr + vi * vi, buf);
  const float i0 = 1.f / (sqrtf(n0) + 1e-30f);
  xr[g] = vr * i0; xi[g] = vi * i0;
  __syncthreads();

  float lambda = 0.f;
  for (int it = 0; it < POWER_ITERS; ++it) {
    float yr = 0.f, yi = 0.f;
    for (int j = 0; j < G; ++j) {
      const float a = Rm[j], b = Im[j];
      const float cr = xr[j], ci = xi[j];
      yr += a * cr - b * ci;
      yi += a * ci + b * cr;
    }
    const float s = blk_sum(yr * yr + yi * yi, buf);  // also syncs readers
    lambda = sqrtf(s);
    const float iv = 1.f / (lambda + 1e-30f);
    xr[g] = yr * iv; xi[g] = yi * iv;
    __syncthreads();
  }
  if (g == 0) {
    const float gamma = 1.f / fmaxf(lambda, 1e-30f);
    ws[OFF_GAMMA + bs] = gamma;
    ws[OFF_THETA + bs] = 1.0f /*ALPHA*/ * ws[OFF_LAM + bs] * gamma;
  }
}

// ================================================================
// Kernel 4: 16 ISTA layers: s = res - g*(DhD@res) + g*Dhr,
//           res = relu(|s| - theta); then min-max normalize final.
// ================================================================
__global__ __launch_bounds__(512) void ista_kernel(
    float* __restrict__ ws, float* __restrict__ out) {
  __shared__ float res[G], buf[512];
  const int bs = blockIdx.x, g = threadIdx.x;
  const float* __restrict__ Rm = ws + OFF_RE + (size_t)bs * G * G + (size_t)g * G;
  const float* __restrict__ Im = ws + OFF_IM + (size_t)bs * G * G + (size_t)g * G;
  const float gamma = ws[OFF_GAMMA + bs];
  const float theta = ws[OFF_THETA + bs];
  const float whr = gamma * ws[OFF_DHR_RE + (size_t)bs * G + g];
  const float whi = gamma * ws[OFF_DHR_IM + (size_t)bs * G + g];
  res[g] = ws[OFF_RES0 + (size_t)bs * G + g];
  __syncthreads();

  float* __restrict__ ia = out + OUT_INIT + (size_t)bs * LAYERS * G;
  float nv = 0.f;
  for (int l = 0; l < LAYERS; ++l) {
    float vr = 0.f, vi = 0.f;
    for (int j = 0; j < G; ++j) {
      const float x = res[j];
      vr += Rm[j] * x;
      vi += Im[j] * x;
    }
    const float sRe = res[g] - gamma * vr + whr;
    const float sIm = -gamma * vi + whi;
    nv = fmaxf(sqrtf(sRe * sRe + sIm * sIm) - theta, 0.f);
    __syncthreads();
    res[g] = nv;
    ia[(size_t)l * G + g] = nv;
    __syncthreads();
  }
  ws[OFF_RESF + (size_t)bs * G + g] = nv;    // raw final for result_ave
  const float lo = blk_min(nv, buf);
  const float hi = blk_max(nv, buf);
  out[(size_t)bs * G + g] = (nv - lo) / (hi - lo + EPSF);
}

// ================================================================
// Kernel 5: result_ave = minmax_norm(mean over S of raw finals)
// ================================================================
__global__ __launch_bounds__(512) void ave_kernel(
    const float* __restrict__ ws, float* __restrict__ out) {
  __shared__ float buf[512];
  const int b = blockIdx.x, g = threadIdx.x;
  float s = 0.f;
  for (int si = 0; si < 9; ++si)
    s += ws[OFF_RESF + ((size_t)(b * 9 + si)) * G + g];
  const float m = s * (1.f / 9.f);
  const float lo = blk_min(m, buf);
  const float hi = blk_max(m, buf);
  out[OUT_AVE + (size_t)b * G + g] = (m - lo) / (hi - lo + EPSF);
}

extern "C" void kernel_launch(void* const* d_in, const int* in_sizes, int n_in,
                              void* d_out, int out_size, void* d_ws, size_t ws_size,
                              hipStream_t stream) {
  (void)in_sizes; (void)n_in; (void)out_size; (void)ws_size;
  const float* Dr = (const float*)d_in[0];
  const float* Di = (const float*)d_in[1];
  const float* Cr = (const float*)d_in[2];
  const float* Ci = (const float*)d_in[3];
  float* out = (float*)d_out;
  float* ws  = (float*)d_ws;

  // 72 bs * 32 i-tiles * 16 j-strips = 36864 wave-tasks; 8 waves/block
  dhd_wmma_kernel<<<4608, 256, 0, stream>>>(Dr, Di, ws);
  setup_kernel<<<BS_TOTAL, 256, 0, stream>>>(Dr, Di, Cr, Ci, ws);
  power_kernel<<<BS_TOTAL, 512, 0, stream>>>(ws);
  ista_kernel<<<BS_TOTAL, 512, 0, stream>>>(ws, out);
  ave_kernel<<<8, 512, 0, stream>>>(ws, out);
}